// FWM_25675314495568
// MI455X (gfx1250) — compile-verified
//
#include <hip/hip_runtime.h>
#include <math.h>

typedef __attribute__((ext_vector_type(2))) float v2f;
typedef __attribute__((ext_vector_type(8))) float v8f;

#define HIDDEN 512
#define QSIZE  256
#define BATCH  4096

// ---------------------------------------------------------------------------
// Kernel A: P[b,n] = sum_k z[b,k] * Wq[n,k] + Wq_b[n]
// One wave per 16x16 tile of P, K-loop in steps of 4 using V_WMMA_F32_16X16X4_F32.
//
// ISA layouts (CDNA5 ISA):
//   A (16x4, MxK):  lane l -> M = l&15 ;  VGPR0 = A[M, 2*(l>>4)], VGPR1 = A[M, 2*(l>>4)+1]
//   B (4x16, KxN):  lane l -> N = l&15 ;  VGPR0 = B[2*(l>>4), N], VGPR1 = B[2*(l>>4)+1, N]
//   C/D (16x16):    lane l -> N = l&15 ;  VGPR r = C[r + 8*(l>>4), N]
// Both A and B per-lane operands are contiguous float2 loads from row-major
// z and Wq respectively (B = Wq^T, so B[k,n] = Wq[n,k]).
// ---------------------------------------------------------------------------
__global__ __launch_bounds__(128) void fwm_proj_wmma(const float* __restrict__ z,
                                                     const float* __restrict__ Wq,
                                                     const float* __restrict__ Wqb,
                                                     float* __restrict__ P) {
    const int lane  = threadIdx.x & 31;
    const int wave  = threadIdx.x >> 5;
    const int tile  = blockIdx.x * 4 + wave;   // 4096 tiles total
    const int tileM = tile >> 4;               // 0..255  (batch rows / 16)
    const int tileN = tile & 15;               // 0..15   (q cols / 16)
    const int row0  = tileM * 16;
    const int col0  = tileN * 16;

    const int laneM = lane & 15;
    const int koff  = (lane >> 4) << 1;        // 0 or 2

    const float* zrow = z  + (size_t)(row0 + laneM) * HIDDEN + koff;
    const float* wrow = Wq + (size_t)(col0 + laneM) * HIDDEN + koff;

    v8f acc = {};
#pragma unroll 8
    for (int k = 0; k < HIDDEN; k += 4) {
        v2f a = *(const v2f*)(zrow + k);
        v2f b = *(const v2f*)(wrow + k);
        // 8 args: (neg_a, A, neg_b, B, c_mod, C, reuse_a, reuse_b)
        acc = __builtin_amdgcn_wmma_f32_16x16x4_f32(false, a, false, b,
                                                    (short)0, acc, false, false);
    }

    const int mHalf = (lane >> 4) << 3;        // 0 or 8
    const float bias = Wqb[col0 + laneM];
#pragma unroll
    for (int r = 0; r < 8; ++r) {
        P[(size_t)(row0 + r + mHalf) * QSIZE + col0 + laneM] = acc[r] + bias;
    }
}

// ---------------------------------------------------------------------------
// Kernel B: per batch row b:
//   e[j]  = elu(P[b,j])+1 = (P>0 ? P+1 : exp(P))
//   S     = sum_j e[j]
//   m     = sigmoid(z[b,:] . Wm + Wm_b)
//   V[b,j] = e[j] * m / S
// One wave per row; 8 rows per 256-thread block.
// ---------------------------------------------------------------------------
__global__ __launch_bounds__(256) void fwm_coeff(const float* __restrict__ z,
                                                 const float* __restrict__ Wm,
                                                 const float* __restrict__ Wmb,
                                                 const float* __restrict__ P,
                                                 float* __restrict__ V) {
    const int lane = threadIdx.x & 31;
    const int wave = threadIdx.x >> 5;
    const int b    = blockIdx.x * 8 + wave;

    // m-logit dot product, lane-strided (coalesced)
    const float* zr = z + (size_t)b * HIDDEN;
    float md = 0.f;
#pragma unroll
    for (int t = 0; t < HIDDEN / 32; ++t) {
        const int j = lane + t * 32;
        md += zr[j] * Wm[j];
    }
#pragma unroll
    for (int off = 16; off > 0; off >>= 1) md += __shfl_xor(md, off, 32);

    // elu+1 over this lane's 8-element slice of the P row (wave covers 1KB row)
    const float* pr = P + (size_t)b * QSIZE + lane * 8;
    float e[8];
    float s = 0.f;
#pragma unroll
    for (int t = 0; t < 8; ++t) {
        const float x = pr[t];
        e[t] = (x > 0.f) ? (x + 1.f) : expf(x);
        s += e[t];
    }
#pragma unroll
    for (int off = 16; off > 0; off >>= 1) s += __shfl_xor(s, off, 32);

    const float m     = 1.f / (1.f + expf(-(md + Wmb[0])));
    const float scale = m / s;

    float* vr = V + (size_t)b * QSIZE + lane * 8;
#pragma unroll
    for (int t = 0; t < 8; ++t) vr[t] = e[t] * scale;
}

// ---------------------------------------------------------------------------
// Kernel C (bandwidth-critical, ~1.07 GB of F streamed once):
//   out[b,i] = sum_j F[b,i,j] * V[b,j]
// One block per batch element, 8 waves x 32 rows. Each lane holds its
// 8-element slice of V in registers; per F row each wave issues two float4
// loads per lane covering a contiguous 1KB row, then a 5-step shuffle
// reduction. 4096 blocks * 8 waves => deep MLP to saturate 23.3 TB/s HBM.
// ---------------------------------------------------------------------------
__global__ __launch_bounds__(256) void fwm_matvec(const float* __restrict__ F,
                                                  const float* __restrict__ V,
                                                  float* __restrict__ out) {
    const int b    = blockIdx.x;
    const int lane = threadIdx.x & 31;
    const int wave = threadIdx.x >> 5;

    const float4* vv = (const float4*)(V + (size_t)b * QSIZE + lane * 8);
    const float4 v0 = vv[0];
    const float4 v1 = vv[1];

    const float* Fb = F + (size_t)b * QSIZE * QSIZE;

#pragma unroll 4
    for (int r = 0; r < 32; ++r) {
        const int i = wave * 32 + r;
        const float4* fr = (const float4*)(Fb + (size_t)i * QSIZE + lane * 8);
        const float4 f0 = fr[0];
        const float4 f1 = fr[1];
        float acc = f0.x * v0.x + f0.y * v0.y + f0.z * v0.z + f0.w * v0.w
                  + f1.x * v1.x + f1.y * v1.y + f1.z * v1.z + f1.w * v1.w;
#pragma unroll
        for (int off = 16; off > 0; off >>= 1) acc += __shfl_xor(acc, off, 32);
        if (lane == 0) out[(size_t)b * QSIZE + i] = acc;
    }
}

// ---------------------------------------------------------------------------
// Launch. Inputs (setup_inputs order): z, F, Wq_w, Wq_b, Wm_w, Wm_b.
// P (4 MB) is staged in d_out (dead storage until kernel C overwrites it,
// sequencing on `stream` guarantees order); V (4 MB) lives in d_ws.
// ---------------------------------------------------------------------------
extern "C" void kernel_launch(void* const* d_in, const int* in_sizes, int n_in,
                              void* d_out, int out_size, void* d_ws, size_t ws_size,
                              hipStream_t stream) {
    const float* z   = (const float*)d_in[0];
    const float* F   = (const float*)d_in[1];
    const float* Wq  = (const float*)d_in[2];
    const float* Wqb = (const float*)d_in[3];
    const float* Wm  = (const float*)d_in[4];
    const float* Wmb = (const float*)d_in[5];

    float* out = (float*)d_out;
    float* P   = (float*)d_out;              // staged projection (overwritten later)
    float* V   = (float*)d_ws;               // 4096*256 floats = 4 MB scratch

    // A: 4096 tiles, 4 waves (tiles) per 128-thread block -> 1024 blocks
    fwm_proj_wmma<<<1024, 128, 0, stream>>>(z, Wq, Wqb, P);
    // B: 8 rows per block -> 512 blocks
    fwm_coeff<<<BATCH / 8, 256, 0, stream>>>(z, Wm, Wmb, P, V);
    // C: one block per batch element
    fwm_matvec<<<BATCH, 256, 0, stream>>>(F, V, out);
}